// PositionalSparseLinear3d_79121887527368
// MI455X (gfx1250) — compile-verified
//
#include <hip/hip_runtime.h>

// Problem constants (from reference)
#define B_   64
#define IN_  (32 * 64 * 64)   // 131072
#define OUT_ (16 * 32 * 32)   // 16384
#define K_   32

typedef __attribute__((ext_vector_type(2))) float v2f;
typedef __attribute__((ext_vector_type(8))) float v8f;

// ---------------------------------------------------------------------------
// Transpose x (B, IN) -> xT (IN, B) so gathers are coalesced across the batch
// dimension (lanes = batch). 64-wide column tiles through LDS.
// grid = IN/64 blocks, block = 256 threads.
// ---------------------------------------------------------------------------
__global__ void __launch_bounds__(256)
psl3d_transpose_kernel(const float* __restrict__ x, float* __restrict__ xT) {
    __shared__ float tile[B_][64 + 1];
    const int i0 = blockIdx.x * 64;
    const int tid = threadIdx.x;
    const int ci = tid & 63;   // column within tile
    const int rb = tid >> 6;   // 0..3

#pragma unroll
    for (int r = 0; r < 16; ++r) {
        const int b = rb + r * 4;
        tile[b][ci] = x[(size_t)b * IN_ + i0 + ci];   // 256B coalesced reads
    }
    __syncthreads();
#pragma unroll
    for (int r = 0; r < 16; ++r) {
        const int i = rb + r * 4;
        const int b = ci;
        xT[(size_t)(i0 + i) * B_ + b] = tile[b][i];   // 256B coalesced writes
    }
}

// ---------------------------------------------------------------------------
// Main kernel: 1 block = 16 outputs x all 64 batches, 4 waves (one 16-batch
// tile per wave). Exact 16(out) x 16(batch) f32 tile accumulated via 128
// block-diagonal v_wmma_f32_16x16x4_f32 steps (same math as FMA, RNE).
//
// MODE 0: xsrc = xT (IN,64) layout  -> gather offset = (idx<<8) + (bb<<2)
// MODE 1: xsrc = x  (64,IN) layout  -> gather offset = (idx<<2) + bb*IN*4
// Connection indices are pre-scaled to byte offsets when staged into LDS so
// the inner loop is: ds_load_b64 (pair) -> v_add -> global_load_b32 (saddr).
// ---------------------------------------------------------------------------
template <int MODE>
__global__ void __launch_bounds__(128)
psl3d_wmma_kernel(const float* __restrict__ xsrc,
                  const int*   __restrict__ conn,
                  const float* __restrict__ weights,
                  float*       __restrict__ out) {
    __shared__ unsigned off_lds[16 * K_];   // pre-scaled byte offsets
    __shared__ float    w_lds[16 * K_];
    __shared__ float    otile[B_][16];

    const int tid    = threadIdx.x;
    const int o_base = blockIdx.x * 16;

    // Stage this group's 16x32 indices (pre-scaled) + weights into LDS, once.
#pragma unroll
    for (int i = tid; i < 16 * K_; i += 128) {
        const unsigned idx = (unsigned)conn[(size_t)o_base * K_ + i];
        off_lds[i] = (MODE == 0) ? (idx << 8) : (idx << 2);
        w_lds[i]   = weights[(size_t)o_base * K_ + i];
    }
    __syncthreads();

    const int lane = tid & 31;
    const int wave = tid >> 5;        // batch tile 0..3
    const int m    = lane & 15;       // A row (output) / B column (batch)
    const int half = lane >> 4;       // 0 or 1 -> K slots {0,1} vs {2,3}
    const int bb   = wave * 16 + m;   // batch element this lane feeds into B
    const unsigned laneoff =
        (MODE == 0) ? ((unsigned)bb << 2) : ((unsigned)bb * (unsigned)(IN_ * 4));
    const char* xbase = (const char*)xsrc;

    v8f C = {0.f, 0.f, 0.f, 0.f, 0.f, 0.f, 0.f, 0.f};

#pragma unroll 8
    for (int kk = 0; kk < (16 * K_) / 4; ++kk) {   // 128 WMMA steps
        const int i0 = kk >> 3;                    // active output row
        const int kc = (kk & 7) * 4 + half * 2;    // K slot pair for this half

        // A(16x4): only row i0 carries weights, other rows are zero.
        const v2f wv = *(const v2f*)&w_lds[m * K_ + kc];   // 8B-aligned pair
        v2f A;
        A.x = (m == i0) ? wv.x : 0.f;
        A.y = (m == i0) ? wv.y : 0.f;

        // B(4x16): gathered x values for output i0's connections kc, kc+1.
        const uint2 oo = *(const uint2*)&off_lds[i0 * K_ + kc];  // ds_load_b64
        v2f Bm;
        Bm.x = *(const float*)(xbase + (oo.x + laneoff));
        Bm.y = *(const float*)(xbase + (oo.y + laneoff));

        // D = A x B + C
        C = __builtin_amdgcn_wmma_f32_16x16x4_f32(
                /*neg_a=*/false, A, /*neg_b=*/false, Bm,
                /*c_mod=*/(short)0, C, /*reuse_a=*/false, /*reuse_b=*/false);
    }

    // D layout: reg r -> output row r (lanes 0-15) / r+8 (lanes 16-31),
    // column N = m -> batch bb. Stage to LDS, then coalesced store.
#pragma unroll
    for (int r = 0; r < 8; ++r) {
        otile[bb][r + half * 8] = C[r];
    }
    __syncthreads();

#pragma unroll
    for (int i = 0; i < 8; ++i) {
        const int idx = i * 128 + tid;
        const int b = idx >> 4;
        const int o = idx & 15;
        out[(size_t)b * OUT_ + o_base + o] = otile[b][o];
    }
}

extern "C" void kernel_launch(void* const* d_in, const int* in_sizes, int n_in,
                              void* d_out, int out_size, void* d_ws, size_t ws_size,
                              hipStream_t stream) {
    const float* x    = (const float*)d_in[0];
    const int*   conn = (const int*)  d_in[1];
    const float* w    = (const float*)d_in[2];
    float*       out  = (float*)d_out;

    const size_t xT_bytes = (size_t)IN_ * B_ * sizeof(float);

    if (ws_size >= xT_bytes) {
        // Transposed (coalesced-gather) path.
        float* xT = (float*)d_ws;
        psl3d_transpose_kernel<<<IN_ / 64, 256, 0, stream>>>(x, xT);
        psl3d_wmma_kernel<0><<<OUT_ / 16, 128, 0, stream>>>(xT, conn, w, out);
    } else {
        // Fallback: gather directly from (B, IN) layout (L2-resident anyway).
        psl3d_wmma_kernel<1><<<OUT_ / 16, 128, 0, stream>>>(x, conn, w, out);
    }
}